// CatNet_76553497084407
// MI455X (gfx1250) — compile-verified
//
#include <hip/hip_runtime.h>
#include <hip/hip_bf16.h>

typedef __attribute__((ext_vector_type(2))) float v2f;
typedef __attribute__((ext_vector_type(8))) float v8f;
typedef unsigned int u32x4 __attribute__((ext_vector_type(4)));
typedef int i32x4 __attribute__((ext_vector_type(4)));
typedef int i32x8 __attribute__((ext_vector_type(8)));

#define T_STEPS 50
#define THETA   0.9999f
#define INV_T   (1.0f / 50.0f)

// Activation buffer geometry (h-major, zero halo row at r=0 and tail rows)
#define L2_ROWS 184   // layer2 input: rows 0..183, data rows 1..180 (hh=r-1), 16 ch
#define L3_ROWS 104   // layer3 input: rows 0..103, data rows 1..87  (hp=r-1), 32 ch

// Exact replica of the reference IF scan (soft reset), returns spike COUNT (0..50).
__device__ __forceinline__ float if_count(float a) {
    float v = 0.0f, cnt = 0.0f;
    #pragma unroll 1
    for (int t = 0; t < T_STEPS; ++t) {
        v += a;
        float s = (v >= THETA) ? 1.0f : 0.0f;
        v -= s * THETA;
        cnt += s;
    }
    return cnt;
}

__device__ __forceinline__ v8f wmma_f32_k4(v2f a, v2f b, v8f c) {
    return __builtin_amdgcn_wmma_f32_16x16x4_f32(
        /*neg_a=*/false, a, /*neg_b=*/false, b,
        /*c_mod=*/(short)0, c, /*reuse_a=*/false, /*reuse_b=*/false);
}

// ---------------------------------------------------------------------------
// Tensor Data Mover: 1-row tensor copy global -> LDS (ndw dwords, ndw < 65536)
// ---------------------------------------------------------------------------
#if __has_builtin(__builtin_amdgcn_tensor_load_to_lds)
#define HAVE_TDM 1
__device__ __forceinline__ void tdm_load_1d(unsigned lds_addr, const void* gptr,
                                            unsigned ndw) {
    unsigned long long ga = (unsigned long long)(uintptr_t)gptr;
    u32x4 g0;
    g0[0] = 1u;                                     // count=1 (valid), user mode
    g0[1] = lds_addr;                               // lds_addr [63:32]
    g0[2] = (unsigned)(ga & 0xffffffffu);           // global_addr [95:64]
    g0[3] = (unsigned)((ga >> 32) & 0x01ffffffu)    // global_addr [120:96]
          | (2u << 30);                             // type=2 ("image") [127:126]
    i32x8 g1;
    g1[0] = (int)(2u << 16);                        // data_size=4B, mask=0, no flags
    g1[1] = (int)((ndw & 0xffffu) << 16);           // tensor_dim0[15:0] @ [63:48]
    g1[2] = (int)(((ndw >> 16) & 0xffffu)           // tensor_dim0[31:16]
          | (1u << 16));                            // tensor_dim1 = 1
    g1[3] = (int)((ndw & 0xffffu) << 16);           // tile_dim0 = ndw @ [127:112]
    g1[4] = 1;                                      // tile_dim1 = 1, tile_dim2 = 0
    g1[5] = (int)ndw;                               // tensor_dim0_stride[31:0]
    g1[6] = 0;
    g1[7] = 0;
    i32x4 z4 = {0, 0, 0, 0};
#if defined(__clang_major__) && __clang_major__ >= 23
    i32x8 z8 = {0, 0, 0, 0, 0, 0, 0, 0};
    __builtin_amdgcn_tensor_load_to_lds(g0, g1, z4, z4, z8, 0);
#else
    __builtin_amdgcn_tensor_load_to_lds(g0, g1, z4, z4, 0);
#endif
}
#else
#define HAVE_TDM 0
#endif

// ---------------------------------------------------------------------------
// Kernel 1: time-sum of x, conv1 (k=3, pad=1, 1->16 ch), IF -> counts
// Output c1t layout: [n][r=184][ic=16], r = h+1; rows 0,181..183 zeroed.
// ---------------------------------------------------------------------------
__global__ __launch_bounds__(128) void k_layer1(
    const float* __restrict__ x,   // [256][1][180][1][50]
    const float* __restrict__ w1,  // [16][1][3][1]
    const float* __restrict__ b1,  // [16]
    float* __restrict__ c1t)       // [256][184][16]
{
    int idx = blockIdx.x * 128 + threadIdx.x;
    if (idx >= 256 * L2_ROWS) return;
    int n = idx / L2_ROWS, r = idx - n * L2_ROWS;
    int h = r - 1;
    float* outp = c1t + (size_t)idx * 16;

    if (h < 0 || h >= 180) {                 // halo rows -> zeros
        #pragma unroll
        for (int oc = 0; oc < 16; ++oc) outp[oc] = 0.0f;
        return;
    }

    const float* xp = x + (size_t)n * 180 * T_STEPS;
    float s[3];
    #pragma unroll
    for (int kh = 0; kh < 3; ++kh) {
        int hh = h + kh - 1;
        float acc = 0.0f;
        if (hh >= 0 && hh < 180) {
            const float* p = xp + (size_t)hh * T_STEPS;
            #pragma unroll 10
            for (int t = 0; t < T_STEPS; ++t) acc += p[t];
        }
        s[kh] = acc;
    }
    #pragma unroll
    for (int oc = 0; oc < 16; ++oc) {
        float a = (s[0] * w1[oc * 3 + 0] + s[1] * w1[oc * 3 + 1] +
                   s[2] * w1[oc * 3 + 2]) * INV_T + b1[oc];
        outp[oc] = if_count(a);
    }
}

// ---------------------------------------------------------------------------
// Kernel 2: conv2 (k=9, pad=1, 16->32) WMMA GEMM M=32 N=174 K=144 (kh-major),
// IF, sum-pool(2)*1.1, IF -> c3t [256][104][32] (h-major + halo). 1 block/sample.
// B[kk][h] = sIn[h*16 + kk]  (linear window thanks to kh-major K + h-major act)
// ---------------------------------------------------------------------------
__global__ __launch_bounds__(128) void k_layer2(
    const float* __restrict__ c1t, // [256][184][16]
    const float* __restrict__ w2,  // [32][16][9][1] flat oc*144 + ic*9 + kh
    const float* __restrict__ b2,  // [32]
    float* __restrict__ c3t)       // [256][104][32]
{
    __shared__ float sIn[L2_ROWS * 16];  // 2944 f: raw copy of c1t[n]
    __shared__ float sW[32 * 144];       // permuted: [oc][kk = kh*16+ic]
    __shared__ float sC2[32 * 176];      // layer2 spike counts, h padded to 176

    const int n = blockIdx.x, tid = threadIdx.x;

    // one-time permuted weight fill (kh-major K)
    for (int i = tid; i < 32 * 144; i += 128) {
        int oc = i / 144, kk = i - oc * 144;
        int kh = kk >> 4, ic = kk & 15;
        sW[i] = w2[oc * 144 + ic * 9 + kh];
    }
    // raw activation slice: TDM async copy (fallback: vector loads)
#if HAVE_TDM
    if (tid < 32) {
        tdm_load_1d((unsigned)(uintptr_t)&sIn[0],
                    c1t + (size_t)n * (L2_ROWS * 16), L2_ROWS * 16);
        __builtin_amdgcn_s_wait_tensorcnt(0);
    }
#else
    for (int i = tid; i < L2_ROWS * 16; i += 128)
        sIn[i] = c1t[(size_t)n * (L2_ROWS * 16) + i];
#endif
    __syncthreads();

    const int wave = tid >> 5, lane = tid & 31;
    const int half = lane >> 4, li = lane & 15;

    // 22 tiles of 16x16 (2 oc-tiles x 11 h-tiles), 4 waves round-robin
    for (int tile = wave; tile < 22; tile += 4) {
        const int ocBase = (tile & 1) * 16;
        const int h      = (tile >> 1) * 16 + li;      // h in 0..175
        const float* Ar = &sW[(ocBase + li) * 144 + 2 * half];
        const float* Br = &sIn[h * 16 + 2 * half];
        v8f acc = {};
        #pragma unroll
        for (int k = 0; k < 144; k += 4) {
            v2f a, b;
            a.x = Ar[k]; a.y = Ar[k + 1];
            b.x = Br[k]; b.y = Br[k + 1];
            acc = wmma_f32_k4(a, b, acc);
        }
        #pragma unroll
        for (int j = 0; j < 8; ++j) {                  // D: vgpr j -> M = j + 8*half
            int oc = ocBase + j + 8 * half;
            sC2[oc * 176 + h] = if_count(acc[j] * INV_T + b2[oc]);
        }
    }
    __syncthreads();

    // sum-pool(2,1)*1.1 on counts, IF, write h-major + halo rows
    for (int i = tid; i < L3_ROWS * 32; i += 128) {
        int r = i >> 5, oc = i & 31;
        float v = 0.0f;
        if (r >= 1 && r <= 87) {
            int hp = r - 1;
            float a3 = 1.1f * (sC2[oc * 176 + 2 * hp] +
                               sC2[oc * 176 + 2 * hp + 1]) * INV_T;
            v = if_count(a3);
        }
        c3t[(size_t)n * (L3_ROWS * 32) + i] = v;
    }
}

// ---------------------------------------------------------------------------
// Kernel 3: conv3 (k=7, pad=1, 32->32) WMMA GEMM M=32 N=83 K=224 (kh-major),
// IF -> c4 [256][32][83] (oc-major for dense readout).
// B[kk][h] = sIn[h*32 + kk]
// ---------------------------------------------------------------------------
__global__ __launch_bounds__(128) void k_layer3(
    const float* __restrict__ c3t, // [256][104][32]
    const float* __restrict__ w3,  // [32][32][7][1] flat oc*224 + ic*7 + kh
    const float* __restrict__ b3,  // [32]
    float* __restrict__ c4)        // [256][32][83]
{
    __shared__ float sIn[L3_ROWS * 32];  // 3328 f: raw copy of c3t[n]
    __shared__ float sW[32 * 224];       // permuted: [oc][kk = kh*32+ic]

    const int n = blockIdx.x, tid = threadIdx.x;

    for (int i = tid; i < 32 * 224; i += 128) {
        int oc = i / 224, kk = i - oc * 224;
        int kh = kk >> 5, ic = kk & 31;
        sW[i] = w3[oc * 224 + ic * 7 + kh];
    }
#if HAVE_TDM
    if (tid < 32) {
        tdm_load_1d((unsigned)(uintptr_t)&sIn[0],
                    c3t + (size_t)n * (L3_ROWS * 32), L3_ROWS * 32);
        __builtin_amdgcn_s_wait_tensorcnt(0);
    }
#else
    for (int i = tid; i < L3_ROWS * 32; i += 128)
        sIn[i] = c3t[(size_t)n * (L3_ROWS * 32) + i];
#endif
    __syncthreads();

    const int wave = tid >> 5, lane = tid & 31;
    const int half = lane >> 4, li = lane & 15;

    // 12 tiles (2 oc-tiles x 6 h-tiles covering 83 cols)
    for (int tile = wave; tile < 12; tile += 4) {
        const int ocBase = (tile & 1) * 16;
        const int h      = (tile >> 1) * 16 + li;      // h in 0..95
        const float* Ar = &sW[(ocBase + li) * 224 + 2 * half];
        const float* Br = &sIn[h * 32 + 2 * half];
        v8f acc = {};
        #pragma unroll
        for (int k = 0; k < 224; k += 4) {
            v2f a, b;
            a.x = Ar[k]; a.y = Ar[k + 1];
            b.x = Br[k]; b.y = Br[k + 1];
            acc = wmma_f32_k4(a, b, acc);
        }
        if (h < 83) {
            #pragma unroll
            for (int j = 0; j < 8; ++j) {
                int oc = ocBase + j + 8 * half;
                c4[((size_t)n * 32 + oc) * 83 + h] =
                    if_count(acc[j] * INV_T + b3[oc]);
            }
        }
    }
}

// ---------------------------------------------------------------------------
// Kernel 4: dense readout: out[n,o] = wf.c4/50 + bf
// ---------------------------------------------------------------------------
__global__ __launch_bounds__(256) void k_dense(
    const float* __restrict__ c4,  // [256][32*83]
    const float* __restrict__ wf,  // [4][32*83]
    const float* __restrict__ bf,  // [4]
    float* __restrict__ out)       // [256][4]
{
    int idx = blockIdx.x * 256 + threadIdx.x;
    if (idx >= 256 * 4) return;
    int n = idx >> 2, o = idx & 3;
    const float* cp = c4 + (size_t)n * 2656;
    const float* wp = wf + (size_t)o * 2656;
    float acc = 0.0f;
    #pragma unroll 4
    for (int i = 0; i < 2656; ++i) acc += cp[i] * wp[i];
    out[n * 4 + o] = acc * INV_T + bf[o];
}

// ---------------------------------------------------------------------------
extern "C" void kernel_launch(void* const* d_in, const int* in_sizes, int n_in,
                              void* d_out, int out_size, void* d_ws, size_t ws_size,
                              hipStream_t stream) {
    const float* x  = (const float*)d_in[0];
    const float* w1 = (const float*)d_in[1];
    const float* b1 = (const float*)d_in[2];
    const float* w2 = (const float*)d_in[3];
    const float* b2 = (const float*)d_in[4];
    const float* w3 = (const float*)d_in[5];
    const float* b3 = (const float*)d_in[6];
    const float* wf = (const float*)d_in[7];
    const float* bf = (const float*)d_in[8];
    float* out = (float*)d_out;

    float* ws  = (float*)d_ws;
    float* c1t = ws;                                     // 256*184*16 = 753664
    float* c3t = c1t + (size_t)256 * L2_ROWS * 16;       // 256*104*32 = 851968
    float* c4  = c3t + (size_t)256 * L3_ROWS * 32;       // 256*32*83  = 679936

    k_layer1<<<(256 * L2_ROWS + 127) / 128, 128, 0, stream>>>(x, w1, b1, c1t);
    k_layer2<<<256, 128, 0, stream>>>(c1t, w2, b2, c3t);
    k_layer3<<<256, 128, 0, stream>>>(c3t, w3, b3, c4);
    k_dense<<<4, 256, 0, stream>>>(c4, wf, bf, out);
}